// WFAConstrainedLoss_84061099917524
// MI455X (gfx1250) — compile-verified
//
#include <hip/hip_runtime.h>
#include <hip/hip_bf16.h>

// ---------------------------------------------------------------------------
// WFA-constrained loss for MI455X (gfx1250).
// Memory-bound (~103 MB read, ~0.07 GFLOP): design is coalesced float4
// streaming -> LDS staging -> per-sample regression reductions done with
// V_WMMA_F32_16X16X4_F32 (A = 16 samples x 4 wl chunk, B = ones), which keeps
// full f32 precision and runs the reductions on the XDL pipe.
// ---------------------------------------------------------------------------

typedef float v2f __attribute__((ext_vector_type(2)));
typedef float v8f __attribute__((ext_vector_type(8)));

#define NWL   112
#define NREG  111          // regression uses wavelengths[0:111]
#define ATM   84
#define SPB   64           // samples per block in wfa kernel
#define TPB_B 128          // 4 waves * 32 lanes; each wave owns 16 samples
#define ROW   113          // padded LDS row stride (gcd(113,64)=1 -> no bank conflicts)
#define TPB_A 256

// ---------------------------------------------------------------------- init
__global__ void init_ws_kernel(float* ws) {
    if (threadIdx.x < 3) ws[threadIdx.x] = 0.0f;
}

// ------------------------------------------------------------ base L1 reduce
__global__ __launch_bounds__(TPB_A)
void base_loss_kernel(const float4* __restrict__ p, const float4* __restrict__ t,
                      float* __restrict__ ws, int n4) {
    int idx    = blockIdx.x * TPB_A + threadIdx.x;
    int stride = gridDim.x * TPB_A;
    float s = 0.0f;
    for (int i = idx; i < n4; i += stride) {
        float4 a = p[i], b = t[i];
        s += fabsf(a.x - b.x);
        s += fabsf(a.y - b.y);
        s += fabsf(a.z - b.z);
        s += fabsf(a.w - b.w);
    }
    // wave32 reduction
    for (int off = 16; off > 0; off >>= 1) s += __shfl_down(s, off, 32);
    __shared__ float part[TPB_A / 32];
    int lane = threadIdx.x & 31;
    int wv   = threadIdx.x >> 5;
    if (lane == 0) part[wv] = s;
    __syncthreads();
    if (wv == 0) {
        s = (lane < (TPB_A / 32)) ? part[lane] : 0.0f;
        for (int off = 4; off > 0; off >>= 1) s += __shfl_down(s, off, 32);
        if (lane == 0) atomicAdd(ws + 0, s);
    }
}

// ------------------------------------------------------- WFA regression body
__global__ __launch_bounds__(TPB_B)
void wfa_kernel(const float* __restrict__ stokes,
                const float* __restrict__ patm,
                const float* __restrict__ wl,
                float* __restrict__ ws) {
    __shared__ float sI[SPB * ROW];
    __shared__ float sV[SPB * ROW];
    __shared__ float rgw[NWL];

    const int tid      = threadIdx.x;
    const int block_s0 = blockIdx.x * SPB;

    // ---- stage I (channel 0) and V (channel 3) rows, coalesced b128 loads
    const int nvec = SPB * (NWL / 4);              // 64 * 28 = 1792, exact /128
    for (int e = tid; e < nvec; e += TPB_B) {
        int s  = e / (NWL / 4);
        int j4 = e % (NWL / 4);
        size_t gbase = (size_t)(block_s0 + s) * 4 * NWL;
        float4 iv = ((const float4*)(stokes + gbase))[j4];
        float4 vv = ((const float4*)(stokes + gbase + 3 * NWL))[j4];
        int b = s * ROW + j4 * 4;
        sI[b + 0] = iv.x; sI[b + 1] = iv.y; sI[b + 2] = iv.z; sI[b + 3] = iv.w;
        sV[b + 0] = vv.x; sV[b + 1] = vv.y; sV[b + 2] = vv.z; sV[b + 3] = vv.w;
    }
    // ---- reciprocal wavelength gradient (from the actual f32 wavelengths)
    if (tid < NWL) {
        float g;
        if (tid == 0)             g = wl[1] - wl[0];
        else if (tid >= NREG - 1) g = wl[NREG - 1] - wl[NREG - 2];  // 110 edge; 111 dummy
        else                      g = 0.5f * (wl[tid + 1] - wl[tid - 1]);
        rgw[tid] = 1.0f / g;
    }
    __syncthreads();

    const int wave    = tid >> 5;
    const int lane    = tid & 31;
    const int m       = lane & 15;      // A-matrix row = sample within 16-tile
    const int khalf   = lane >> 4;      // lanes 16-31 hold K = +2,+3 of each chunk
    const int s_local = wave * 16 + m;
    const float* myI  = &sI[s_local * ROW];
    const float* myV  = &sV[s_local * ROW];

    v8f c_sx = {}, c_sy = {}, c_sxx = {}, c_sxy = {};
    const v2f bones = {1.0f, 1.0f};     // B = ones(4x16) -> D rows = running sums

    #pragma unroll
    for (int chunk = 0; chunk < 28; ++chunk) {
        const int kbase = chunk * 4 + khalf * 2;
        float d[2], v[2];
        #pragma unroll
        for (int t = 0; t < 2; ++t) {
            int j  = kbase + t;                         // 0..111 (111 = zero pad)
            int jc = (j < NREG) ? j : (NREG - 1);
            int jm = (jc > 0) ? (jc - 1) : 0;
            int jp = (jc < NREG - 1) ? (jc + 1) : (NREG - 1);
            float sc   = (jp - jm == 2) ? 0.5f : 1.0f;  // central vs one-sided
            float pred = (j < NREG) ? 1.0f : 0.0f;
            float dI   = (myI[jp] - myI[jm]) * sc * rgw[jc] * pred;
            d[t] = dI;
            v[t] = myV[jc] * pred;
        }
        v2f a_d  = { d[0],        d[1]        };
        v2f a_v  = { v[0],        v[1]        };
        v2f a_d2 = { d[0] * d[0], d[1] * d[1] };
        v2f a_dv = { d[0] * v[0], d[1] * v[1] };
        c_sx  = __builtin_amdgcn_wmma_f32_16x16x4_f32(false, a_d,  false, bones, (short)0, c_sx,  false, false);
        c_sy  = __builtin_amdgcn_wmma_f32_16x16x4_f32(false, a_v,  false, bones, (short)0, c_sy,  false, false);
        c_sxx = __builtin_amdgcn_wmma_f32_16x16x4_f32(false, a_d2, false, bones, (short)0, c_sxx, false, false);
        c_sxy = __builtin_amdgcn_wmma_f32_16x16x4_f32(false, a_dv, false, bones, (short)0, c_sxy, false, false);
    }

    // D layout: VGPR r -> M=r (lanes 0-15) / M=8+r (lanes 16-31); all columns
    // identical since B==1. Lanes 0 and 16 each read out 8 samples' quadruples.
    if (m == 0) {
        const float n    = (float)NREG;
        const float KDEN = (float)(4.6686e-13 * 6301.5 * 6301.5 * 1.5);
        float msum = 0.0f, dsum = 0.0f;
        #pragma unroll
        for (int r = 0; r < 8; ++r) {
            float sx = c_sx[r], sy = c_sy[r], sxx = c_sxx[r], sxy = c_sxy[r];
            float coef = (n * sxy - sx * sy) / (n * sxx - sx * sx);
            float wb   = -coef / KDEN;
            float awb  = fabsf(wb);
            float mask = (awb < 1.0e6f) ? 1.0f : 0.0f;   // NaN -> 0, matches jnp
            int gs = block_s0 + wave * 16 + khalf * 8 + r;
            const float* pa = patm + (size_t)gs * ATM;
            float pb   = (pa[3] + pa[7] + pa[11]) / 3.0f;  // mean of B_los at itau 0..2
            float lp   = log10f(fabsf(pb) + 1e-10f);
            float lw   = log10f(awb + 1e-10f);
            msum += mask;
            dsum += mask * fabsf(lp - lw);
        }
        atomicAdd(ws + 1, msum);
        atomicAdd(ws + 2, dsum);
    }
}

// ------------------------------------------------------------------ finalize
__global__ void finalize_kernel(const float* __restrict__ ws,
                                float* __restrict__ out, float n_base) {
    if (threadIdx.x == 0 && blockIdx.x == 0) {
        float base     = ws[0] / n_base;
        float cnt      = ws[1];
        float wfa_mean = ws[2] / fmaxf(cnt, 1.0f);
        bool  apply    = (base < 10.0f) && (cnt > 0.0f);
        out[0] = apply ? (0.5f * base + 0.5f * wfa_mean) : base;
        out[1] = base;
        out[2] = apply ? wfa_mean : 0.0f;
    }
}

// -------------------------------------------------------------------- launch
extern "C" void kernel_launch(void* const* d_in, const int* in_sizes, int n_in,
                              void* d_out, int out_size, void* d_ws, size_t ws_size,
                              hipStream_t stream) {
    const float* stokes = (const float*)d_in[0];   // (B, 4, 112) f32
    const float* patm   = (const float*)d_in[1];   // (B, 84) f32
    const float* tatm   = (const float*)d_in[2];   // (B, 84) f32
    const float* wl     = (const float*)d_in[3];   // (112,) f32
    float* out = (float*)d_out;
    float* ws  = (float*)d_ws;

    const int batch = in_sizes[1] / ATM;           // 65536
    const int n_elems = batch * ATM;

    init_ws_kernel<<<1, 32, 0, stream>>>(ws);
    base_loss_kernel<<<1024, TPB_A, 0, stream>>>((const float4*)patm,
                                                 (const float4*)tatm,
                                                 ws, n_elems / 4);
    wfa_kernel<<<batch / SPB, TPB_B, 0, stream>>>(stokes, patm, wl, ws);
    finalize_kernel<<<1, 32, 0, stream>>>(ws, out, (float)n_elems);
}